// YOLOLoss_49392123904082
// MI455X (gfx1250) — compile-verified
//
#include <hip/hip_runtime.h>

// ---------------- problem constants ----------------
#define GSZ   52
#define NB    32
#define GG    (GSZ*GSZ)        // 2704
#define NCH   255              // 3*(5+80)
#define NCLS  80
#define NANCH 3
#define NBOX  2
#define Q4    (GG/4)           // 676 float4 per conf plane
#define TOT4  (NB*Q4)          // 21632
#define T1    256
#define NBLK1 85               // ceil(21632/256)

typedef __attribute__((ext_vector_type(2))) float v2f;
typedef __attribute__((ext_vector_type(8))) float v8f;

// ---- wave32 sum via V_WMMA_F32_16X16X4_F32 with B = ones ----
// A layout (16x4 f32): lanes 0-15 -> A[m][0]=v (VGPR0), A[m][1]=0 (VGPR1);
// lanes 16-31 -> A[m][2]=v, A[m][3]=0.  With B all-ones:
// D[m][n] = v(lane m) + v(lane m+16)  =>  sum over lane's 8 D regs gives a
// half-row-range partial; xor-16 exchange completes the 32-lane total.
__device__ __forceinline__ float wave_sum_wmma(float v) {
  v2f a; a[0] = v;    a[1] = 0.0f;
  v2f b; b[0] = 1.0f; b[1] = 1.0f;
  v8f c = {};
  v8f d = __builtin_amdgcn_wmma_f32_16x16x4_f32(
      /*neg_a=*/false, a, /*neg_b=*/false, b,
      /*c_mod=*/(short)0, c, /*reuse_a=*/false, /*reuse_b=*/false);
  float s = d[0] + d[1] + d[2] + d[3] + d[4] + d[5] + d[6] + d[7];
  s += __shfl_xor(s, 16, 32);
  return s;  // total in every lane
}

__device__ __forceinline__ float sigmoidf(float x) {
  return 1.0f / (1.0f + expf(-x));
}
__device__ __forceinline__ float bce_t0(float p) {   // BCE(p, 0)
  return -fmaxf(logf(1.0f - p), -100.0f);
}
__device__ __forceinline__ float bce_t1(float p) {   // BCE(p, 1)
  return -fmaxf(logf(p), -100.0f);
}

// ---------------- kernel 1: noobj BCE over the conf-0 plane ----------------
__global__ void __launch_bounds__(T1)
noobj_partial_kernel(const float* __restrict__ pred, float* __restrict__ part) {
  __shared__ float s_w[T1 / 32];
  int idx = blockIdx.x * T1 + threadIdx.x;
  float v = 0.0f;
  if (idx < TOT4) {
    int b = idx / Q4;
    int r = idx - b * Q4;
    const float4* p4 =
        (const float4*)(pred + ((size_t)b * NCH + 4) * GG);
    float4 p = p4[r];
    v  = bce_t0(p.x);
    v += bce_t0(p.y);
    v += bce_t0(p.z);
    v += bce_t0(p.w);
  }
  float s = wave_sum_wmma(v);                 // full wave active here
  int wv = threadIdx.x >> 5, ln = threadIdx.x & 31;
  if (ln == 0) s_w[wv] = s;
  __syncthreads();
  if (threadIdx.x == 0) {
    float acc = 0.0f;
    for (int i = 0; i < T1 / 32; ++i) acc += s_w[i];
    part[blockIdx.x] = acc;                   // deterministic, no atomics
  }
}

// ---------------- kernel 2: object cells + final combine ----------------
__global__ void __launch_bounds__(64)
yolo_final_kernel(const float* __restrict__ pred,
                  const float* __restrict__ bbx,
                  const float* __restrict__ anch,
                  const int*   __restrict__ lab,
                  const float* __restrict__ part,
                  float* __restrict__ out) {
  __shared__ int   s_ci[64], s_cj[64];
  __shared__ float s_acc[2][5];

  const int t = threadIdx.x;       // 64 threads = (batch, box)
  const int b = t >> 1, n = t & 1;
  const float cell = 1.0f / (float)GSZ;

  const float* bb = bbx + (size_t)(b * NBOX + n) * 4;
  float bw = bb[2], bh = bb[3];
  float cxf = bb[0] + bw * 0.5f;
  float cyf = bb[1] + bh * 0.5f;
  int ci = (int)floorf(cxf / cell);
  int cj = (int)floorf(cyf / cell);
  s_ci[t] = ci; s_cj[t] = cj;
  __syncthreads();
  // reference scatter is last-write-wins: box 0 yields to box 1 on collision
  bool owner = (n == 1) || (s_ci[t + 1] != ci) || (s_cj[t + 1] != cj);

  float offx = (float)ci * cell, offy = (float)cj * cell;
  const float eps = 1e-8f;
  float tx = -logf(1.0f / (cxf - offx + eps) - 1.0f);
  float ty = -logf(1.0f / (cyf - offy + eps) - 1.0f);
  float btx = sigmoidf(tx) + offx;
  float bty = sigmoidf(ty) + offy;

  size_t sp = (size_t)b * NCH * GG + (size_t)cj * GSZ + ci;
  int lbl = lab[b * NBOX + n];

  int best = 0; float bestiou = -1e30f;
  float p_raw[NANCH][4], twx_a[NANCH], twy_a[NANCH];
  for (int a = 0; a < NANCH; ++a) {
    float aw = anch[2 * a], ah = anch[2 * a + 1];
    float twx = logf(bw / aw), twy = logf(bh / ah);
    twx_a[a] = twx; twy_a[a] = twy;
    float q0 = pred[sp + (size_t)(a * 85 + 0) * GG];
    float q1 = pred[sp + (size_t)(a * 85 + 1) * GG];
    float q2 = pred[sp + (size_t)(a * 85 + 2) * GG];
    float q3 = pred[sp + (size_t)(a * 85 + 3) * GG];
    p_raw[a][0] = q0; p_raw[a][1] = q1; p_raw[a][2] = q2; p_raw[a][3] = q3;
    float px = sigmoidf(q0) + offx, py = sigmoidf(q1) + offy;
    float pw = aw * expf(q2),       ph = ah * expf(q3);
    float btw = aw * expf(twx),     bth = ah * expf(twy);  // match ref rounding
    float ltx = fmaxf(px - pw * 0.5f, btx - btw * 0.5f);
    float lty = fmaxf(py - ph * 0.5f, bty - bth * 0.5f);
    float rbx = fminf(px + pw * 0.5f, btx + btw * 0.5f);
    float rby = fminf(py + ph * 0.5f, bty + bth * 0.5f);
    float wi = fmaxf(rbx - ltx, 0.0f), hi = fmaxf(rby - lty, 0.0f);
    float inter = wi * hi;
    float uni = pw * ph + btw * bth - inter;
    float iou = inter / (uni + 1e-8f);
    if (iou > bestiou) { bestiou = iou; best = a; }  // first-max tie rule
  }

  float coord = 0.0f, obj = 0.0f, cls = 0.0f, corr = 0.0f, cnt = 0.0f;
  if (owner) {
    cnt = 1.0f;
    float dx = p_raw[best][0] - tx;
    float dy = p_raw[best][1] - ty;
    float dw = p_raw[best][2] - twx_a[best];
    float dh = p_raw[best][3] - twy_a[best];
    coord = dx * dx + dy * dy + dw * dw + dh * dh;
    obj = bce_t1(pred[sp + (size_t)(best * 85 + 4) * GG]);
    for (int c = 0; c < NCLS; ++c) {
      float p = pred[sp + (size_t)(best * 85 + 5 + c) * GG];
      cls += (c == lbl) ? bce_t1(p) : bce_t0(p);
    }
    // this cell is masked out of the noobj sum (m==1)
    corr = bce_t0(pred[sp + (size_t)4 * GG]);
  }

  // full-wave WMMA reductions (all 64 lanes re-converged here)
  float rc = wave_sum_wmma(coord);
  float ro = wave_sum_wmma(obj);
  float rk = wave_sum_wmma(cls);
  float rr = wave_sum_wmma(corr);
  float rn = wave_sum_wmma(cnt);
  int wv = t >> 5, ln = t & 31;
  if (ln == 0) {
    s_acc[wv][0] = rc; s_acc[wv][1] = ro; s_acc[wv][2] = rk;
    s_acc[wv][3] = rr; s_acc[wv][4] = rn;
  }
  __syncthreads();
  if (t == 0) {
    float coord_s = s_acc[0][0] + s_acc[1][0];
    float obj_s   = s_acc[0][1] + s_acc[1][1];
    float cls_s   = s_acc[0][2] + s_acc[1][2];
    float corr_s  = s_acc[0][3] + s_acc[1][3];
    float nobj    = s_acc[0][4] + s_acc[1][4];
    float psum = 0.0f;
    for (int i = 0; i < NBLK1; ++i) psum += part[i];
    float n_noobj = (float)(NB * GG) - nobj;
    float loss = coord_s / (nobj * 4.0f)
               + obj_s   / nobj
               + cls_s   / (nobj * (float)NCLS)
               + (psum - corr_s) / n_noobj;
    out[0] = loss;
  }
}

extern "C" void kernel_launch(void* const* d_in, const int* in_sizes, int n_in,
                              void* d_out, int out_size, void* d_ws, size_t ws_size,
                              hipStream_t stream) {
  const float* pred = (const float*)d_in[0];
  const float* bbx  = (const float*)d_in[1];
  const float* anch = (const float*)d_in[2];
  const int*   lab  = (const int*)d_in[3];
  float* part = (float*)d_ws;          // NBLK1 floats of scratch
  float* out  = (float*)d_out;

  noobj_partial_kernel<<<NBLK1, T1, 0, stream>>>(pred, part);
  yolo_final_kernel<<<1, 64, 0, stream>>>(pred, bbx, anch, lab, part, out);
}